// DynamicGraphModel_50921132261402
// MI455X (gfx1250) — compile-verified
//
#include <hip/hip_runtime.h>
#include <hip/hip_bf16.h>

typedef float v2f __attribute__((ext_vector_type(2)));
typedef float v8f __attribute__((ext_vector_type(8)));

#define FINF __builtin_inff()

// low 32 bits of a generic pointer into LDS aperture == LDS byte offset
__device__ __forceinline__ unsigned lds_off(const void* p) {
  return (unsigned)(uintptr_t)p;
}

// ---------------------------------------------------------------------------
// Kernel A: row squared norms
// ---------------------------------------------------------------------------
template<int D>
__global__ __launch_bounds__(256) void row_norm_kernel(const float* __restrict__ X,
                                                       float* __restrict__ sq, int N) {
  int i = blockIdx.x * blockDim.x + threadIdx.x;
  if (i >= N) return;
  const float* r = X + (size_t)i * D;
  float s = 0.0f;
#pragma unroll 8
  for (int d = 0; d < D; ++d) s += r[d] * r[d];
  sq[i] = s;
}

// ---------------------------------------------------------------------------
// Kernel B: fused Gram-matrix (fp32 WMMA) + streaming top-k kNN.
// One wave (32 threads) per 16-row block; loops over all column tiles of 16.
//   column tile staged via GLOBAL_LOAD_ASYNC_TO_LDS_B128 (ASYNCcnt)
//   S = X_rows (16xD) @ X_cols^T (Dx16) via V_WMMA_F32_16X16X4_F32,
//       two independent accumulators for ILP
//   d2 = |i|^2 + |j|^2 - 2 S,   self -> +inf,   per-row sorted top-K in regs
// ---------------------------------------------------------------------------
template<int D, int KNN>
__global__ __launch_bounds__(32) void knn_kernel(const float* __restrict__ X,
                                                 const float* __restrict__ sq,
                                                 int* __restrict__ idx_out, int N) {
  static_assert(D % 8 == 0, "D multiple of 8");
  constexpr int LDB = D + 4;                 // padded LDS row stride (floats)
  __shared__ float ldsB[16 * LDB];           // staged column-tile rows
  __shared__ float ldsD[16 * 17];            // 16x16 distance tile
  __shared__ float ldsMD[32 * KNN];          // merge buffers
  __shared__ int   ldsMI[32 * KNN];

  const int lane = threadIdx.x;              // 0..31 (wave32)
  const int l15  = lane & 15;
  const int half = lane >> 4;                // 0 or 1 (selects K-pair)
  const int rowBase = blockIdx.x * 16;

  // ---- load persistent A fragments: lane holds X[rowBase+l15][4t+2*half+{0,1}]
  v2f afrag[D / 4];
  {
    int r = rowBase + l15; if (r > N - 1) r = N - 1;
    const float* arow = X + (size_t)r * D;
#pragma unroll
    for (int t = 0; t < D / 4; ++t) {
      afrag[t].x = arow[4 * t + 2 * half + 0];
      afrag[t].y = arow[4 * t + 2 * half + 1];
    }
  }
  // ---- squared norms of my 8 C-rows (lanes 0-15 -> M 0..7, 16-31 -> M 8..15)
  float sqA[8];
#pragma unroll
  for (int r = 0; r < 8; ++r) {
    int rr = rowBase + half * 8 + r; if (rr > N - 1) rr = N - 1;
    sqA[r] = sq[rr];
  }

  // ---- per-lane sorted top-K (ascending distance)
  float bestD[KNN]; int bestI[KNN];
#pragma unroll
  for (int j = 0; j < KNN; ++j) { bestD[j] = FINF; bestI[j] = 0; }

  // per-lane fixed staging assignment (lane covers ITERS float4 chunks)
  constexpr int CHUNKS_PER_ROW = D / 4;
  constexpr int ITERS = (16 * CHUNKS_PER_ROW) / 32;

  for (int colBase = 0; colBase < N; colBase += 16) {
    // -- async memory->LDS staging of the 16 column rows (no VGPR round trip)
#pragma unroll
    for (int it = 0; it < ITERS; ++it) {
      int chunk = it * 32 + lane;
      int r = chunk / CHUNKS_PER_ROW;
      int c = chunk % CHUNKS_PER_ROW;
      int gr = colBase + r; if (gr > N - 1) gr = N - 1;
      unsigned long long ga = (unsigned long long)(uintptr_t)(X + (size_t)gr * D + c * 4);
      unsigned dst = lds_off(&ldsB[r * LDB + c * 4]);
      asm volatile("global_load_async_to_lds_b128 %0, %1, off"
                   :: "v"(dst), "v"(ga) : "memory");
    }
    asm volatile("s_wait_asynccnt 0x0" ::: "memory");
    __syncthreads();

    // -- preload all B fragments for this tile (batched ds_loads, one wait)
    v2f bfrag[D / 4];
#pragma unroll
    for (int t = 0; t < D / 4; ++t)
      bfrag[t] = *(const v2f*)&ldsB[l15 * LDB + 4 * t + 2 * half];

    // -- 16x16 fp32 Gram tile: two independent WMMA accumulator chains
    v8f c0 = {}, c1 = {};
#pragma unroll
    for (int t = 0; t < D / 4; t += 2) {
      c0 = __builtin_amdgcn_wmma_f32_16x16x4_f32(
          false, afrag[t],     false, bfrag[t],     (short)0, c0, false, false);
      c1 = __builtin_amdgcn_wmma_f32_16x16x4_f32(
          false, afrag[t + 1], false, bfrag[t + 1], (short)0, c1, false, false);
    }

    // -- convert to squared distances, exclude self / OOB, drop tile to LDS
    float sqB = sq[(colBase + l15 > N - 1) ? (N - 1) : (colBase + l15)];
    int gcol = colBase + l15;
#pragma unroll
    for (int r = 0; r < 8; ++r) {
      int m = half * 8 + r;
      float d = sqA[r] + sqB - 2.0f * (c0[r] + c1[r]);
      if (gcol == rowBase + m || gcol >= N || rowBase + m >= N) d = FINF;
      ldsD[m * 17 + l15] = d;
    }
    __syncthreads();

    // -- top-K scan: lane (l15,half) owns row l15, columns half*8..half*8+7
#pragma unroll
    for (int j = 0; j < 8; ++j) {
      int n = half * 8 + j;
      float d = ldsD[l15 * 17 + n];
      if (d < bestD[KNN - 1]) {
        bestD[KNN - 1] = d; bestI[KNN - 1] = colBase + n;
#pragma unroll
        for (int q = KNN - 1; q > 0; --q) {
          if (bestD[q] < bestD[q - 1]) {
            float td = bestD[q]; bestD[q] = bestD[q - 1]; bestD[q - 1] = td;
            int   ti = bestI[q]; bestI[q] = bestI[q - 1]; bestI[q - 1] = ti;
          }
        }
      }
    }
    __syncthreads();   // before next tile overwrites ldsB/ldsD
  }

  // ---- merge the two half-lane lists per row, lane<16 writes result
#pragma unroll
  for (int j = 0; j < KNN; ++j) {
    ldsMD[lane * KNN + j] = bestD[j];
    ldsMI[lane * KNN + j] = bestI[j];
  }
  __syncthreads();
  if (half == 0 && rowBase + l15 < N) {
#pragma unroll
    for (int j = 0; j < KNN; ++j) {
      float d = ldsMD[(lane + 16) * KNN + j];
      int  gi = ldsMI[(lane + 16) * KNN + j];
      if (d < bestD[KNN - 1]) {
        bestD[KNN - 1] = d; bestI[KNN - 1] = gi;
#pragma unroll
        for (int q = KNN - 1; q > 0; --q) {
          if (bestD[q] < bestD[q - 1]) {
            float td = bestD[q]; bestD[q] = bestD[q - 1]; bestD[q - 1] = td;
            int   ti = bestI[q]; bestI[q] = bestI[q - 1]; bestI[q - 1] = ti;
          }
        }
      }
    }
    int row = rowBase + l15;
#pragma unroll
    for (int j = 0; j < KNN; ++j) idx_out[(size_t)row * KNN + j] = bestI[j];
  }
}

// ---------------------------------------------------------------------------
// Kernel C: EdgeConv  out[n][h] = max_k ( W[h]·(x_{idx[n][k]} - x_n) + b[h] )
// one block (H threads) per node; edge features staged in LDS (broadcast reads)
// ---------------------------------------------------------------------------
template<int KNN, int D, int H>
__global__ __launch_bounds__(H) void edge_conv_kernel(const float* __restrict__ X,
                                                      const int* __restrict__ idx,
                                                      const float* __restrict__ W,
                                                      const float* __restrict__ b,
                                                      float* __restrict__ out, int N) {
  __shared__ float e[KNN][D];
  int n = blockIdx.x;
  int t = threadIdx.x;                       // output channel
  const float* xi = X + (size_t)n * D;
  for (int kk = 0; kk < KNN; ++kk) {
    int j = idx[(size_t)n * KNN + kk];
    if (j < 0) j = 0; if (j > N - 1) j = N - 1;
    const float* xj = X + (size_t)j * D;
    for (int d = t; d < D; d += H) e[kk][d] = xj[d] - xi[d];
  }
  __syncthreads();

  const float4* w4 = (const float4*)(W + (size_t)t * D);
  float acc[KNN];
#pragma unroll
  for (int kk = 0; kk < KNN; ++kk) acc[kk] = 0.0f;
#pragma unroll 2
  for (int d4 = 0; d4 < D / 4; ++d4) {
    float4 wv = w4[d4];
#pragma unroll
    for (int kk = 0; kk < KNN; ++kk) {
      float4 ev = ((const float4*)e[kk])[d4];
      acc[kk] = fmaf(wv.x, ev.x, fmaf(wv.y, ev.y, fmaf(wv.z, ev.z, fmaf(wv.w, ev.w, acc[kk]))));
    }
  }
  float best = acc[0];
#pragma unroll
  for (int kk = 1; kk < KNN; ++kk) best = fmaxf(best, acc[kk]);
  out[(size_t)n * H + t] = best + b[t];
}

// ---------------------------------------------------------------------------
// Kernel D: classifier head  out[n][c] = h2[n]·Wl[c] + bl[c]   (C=10, H=128)
// ---------------------------------------------------------------------------
__global__ __launch_bounds__(256) void final_linear_kernel(const float* __restrict__ h2,
                                                           const float* __restrict__ Wl,
                                                           const float* __restrict__ bl,
                                                           float* __restrict__ out, int N) {
  int i = blockIdx.x * blockDim.x + threadIdx.x;
  if (i >= N * 10) return;
  int n = i / 10, c = i % 10;
  const float4* hv = (const float4*)(h2 + (size_t)n * 128);
  const float4* wv = (const float4*)(Wl + (size_t)c * 128);
  float acc = 0.0f;
#pragma unroll 4
  for (int q = 0; q < 32; ++q) {
    float4 h = hv[q], w = wv[q];
    acc = fmaf(h.x, w.x, fmaf(h.y, w.y, fmaf(h.z, w.z, fmaf(h.w, w.w, acc))));
  }
  out[i] = acc + bl[c];
}

// ---------------------------------------------------------------------------
extern "C" void kernel_launch(void* const* d_in, const int* in_sizes, int n_in,
                              void* d_out, int out_size, void* d_ws, size_t ws_size,
                              hipStream_t stream) {
  const float* x  = (const float*)d_in[0];
  const float* W1 = (const float*)d_in[1];
  const float* b1 = (const float*)d_in[2];
  const float* W2 = (const float*)d_in[3];
  const float* b2 = (const float*)d_in[4];
  const float* Wl = (const float*)d_in[5];
  const float* bl = (const float*)d_in[6];
  float* out = (float*)d_out;

  const int N = in_sizes[0] / 64;            // 10000
  const int NT = (N + 15) / 16;

  auto align256 = [](size_t v) { return (v + 255) & ~(size_t)255; };
  char* ws = (char*)d_ws;
  float* sq   = (float*)ws;  ws += align256((size_t)N * sizeof(float));
  int*   idx1 = (int*)ws;    ws += align256((size_t)N * 5 * sizeof(int));
  float* h1   = (float*)ws;  ws += align256((size_t)N * 128 * sizeof(float));
  int*   idx2 = (int*)ws;    ws += align256((size_t)N * 10 * sizeof(int));
  float* h2   = (float*)ws;

  // --- layer 1: kNN(k=5) on x (D=64) + EdgeConv -> h1[N,128]
  row_norm_kernel<64><<<(N + 255) / 256, 256, 0, stream>>>(x, sq, N);
  knn_kernel<64, 5><<<NT, 32, 0, stream>>>(x, sq, idx1, N);
  edge_conv_kernel<5, 64, 128><<<N, 128, 0, stream>>>(x, idx1, W1, b1, h1, N);

  // --- layer 2: kNN(k=10) on h1 (D=128) + EdgeConv -> h2[N,128]
  row_norm_kernel<128><<<(N + 255) / 256, 256, 0, stream>>>(h1, sq, N);
  knn_kernel<128, 10><<<NT, 32, 0, stream>>>(h1, sq, idx2, N);
  edge_conv_kernel<10, 128, 128><<<N, 128, 0, stream>>>(h1, idx2, W2, b2, h2, N);

  // --- head
  final_linear_kernel<<<(N * 10 + 255) / 256, 256, 0, stream>>>(h2, Wl, bl, out, N);
}